// InstanceBank_18476949307682
// MI455X (gfx1250) — compile-verified
//
#include <hip/hip_runtime.h>
#include <hip/hip_bf16.h>
#include <math.h>
#include <stdint.h>

#define CHANNEL   256
#define MEM_SIZE  684
#define POS_NUM   128
#define NEG_NUM   512
#define CLASS_NUM 60
#define BANK_ROWS (MEM_SIZE * CLASS_NUM)   /* 41040 */
#define BATCH     512
#define MS_TILES  (BATCH / 32)             /* 16 super-tiles of 32 rows  */
#define NS_TILES  ((BANK_ROWS + 31) / 32)  /* 1283 super-tiles of 32 cols (last is half) */

typedef __attribute__((ext_vector_type(2))) float v2f;
typedef __attribute__((ext_vector_type(8))) float v8f;

// ---------------------------------------------------------------- utilities
__device__ __forceinline__ unsigned fkey(float x) {
  // monotonic float->uint key (ascending)
  unsigned u = __float_as_uint(x);
  return (u & 0x80000000u) ? ~u : (u | 0x80000000u);
}
__device__ __forceinline__ float keyToFloat(unsigned k) {
  unsigned u = (k & 0x80000000u) ? (k & 0x7FFFFFFFu) : ~k;
  return __uint_as_float(u);
}

// ---------------------------------------------------------------- tiny setup
__global__ void zero_out_k(float* out) { out[0] = 0.0f; }

__global__ void flag_init_k(const float* __restrict__ bank_flag,
                            float* __restrict__ flagUpd) {
  int m = blockIdx.x * blockDim.x + threadIdx.x;
  if (m < BANK_ROWS) flagUpd[m] = bank_flag[m];
}

__global__ void flag_scatter_k(const int* __restrict__ enq,
                               float* __restrict__ flagUpd) {
  int t = threadIdx.x;
  if (t < BATCH) flagUpd[enq[t]] = 1.0f;
}

// ---------------------------------------------------------------- WMMA GEMM
// all_pairs[n][m] = dot(f[n], bank[m]); M=512, N=41040, K=256, fp32.
// Each wave computes a 32x32 (2x2 tile) block with V_WMMA_F32_16X16X4_F32:
// A/B loaded once per K-step and reused across the 2x2 tile grid, so the
// load:wmma instruction ratio is 1:1 and L2 traffic is 16KB/tile (vs 32KB
// for the naive 1-tile-per-wave scheme). Last N super-column is 32x16.
// Layouts (ISA 7.12.2): A 16x4 f32: lanes 0-15 -> M rows, vgprs=K{0,1};
// lanes 16-31 -> K{2,3}. B mirrored over N. C: M = v + 8*(lane>=16), N=lane&15.
template <bool FULLN>
__device__ __forceinline__ void gemm_tiles(const float* __restrict__ aP0,
                                           const float* __restrict__ aP1,
                                           const float* __restrict__ bP0,
                                           const float* __restrict__ bP1,
                                           v8f& c00, v8f& c01, v8f& c10, v8f& c11) {
#pragma unroll 4
  for (int k = 0; k < CHANNEL; k += 4) {
    v2f a0 = *(const v2f*)(aP0 + k);
    v2f a1 = *(const v2f*)(aP1 + k);
    v2f b0 = *(const v2f*)(bP0 + k);
    c00 = __builtin_amdgcn_wmma_f32_16x16x4_f32(false, a0, false, b0, (short)0, c00, false, false);
    c10 = __builtin_amdgcn_wmma_f32_16x16x4_f32(false, a1, false, b0, (short)0, c10, false, false);
    if (FULLN) {
      v2f b1 = *(const v2f*)(bP1 + k);
      c01 = __builtin_amdgcn_wmma_f32_16x16x4_f32(false, a0, false, b1, (short)0, c01, false, false);
      c11 = __builtin_amdgcn_wmma_f32_16x16x4_f32(false, a1, false, b1, (short)0, c11, false, false);
    }
  }
}

__global__ __launch_bounds__(256)
void gemm_wmma_k(const float* __restrict__ f, const float* __restrict__ bank,
                 float* __restrict__ ap) {
  int gwave  = (blockIdx.x * 256 + (int)threadIdx.x) >> 5;
  int lane   = threadIdx.x & 31;
  int mSuper = gwave & (MS_TILES - 1);   // 16 super-tiles -> & 15
  int nSuper = gwave >> 4;
  if (nSuper >= NS_TILES) return;        // grid exact; wave-uniform guard

  int l16  = lane & 15;
  int half = lane >> 4;                  // 0: K+0..1, 1: K+2..3
  int mBase = mSuper * 32;
  int nBase = nSuper * 32;
  bool fullN = (nBase + 32) <= BANK_ROWS;  // wave-uniform (EXEC stays all-1s)

  const float* aP0 = f    + (size_t)(mBase + l16) * CHANNEL + half * 2;
  const float* aP1 = aP0  + 16 * CHANNEL;
  const float* bP0 = bank + (size_t)(nBase + l16) * CHANNEL + half * 2;
  const float* bP1 = bP0  + 16 * CHANNEL;

  v8f c00 = {}, c01 = {}, c10 = {}, c11 = {};
  if (fullN) gemm_tiles<true >(aP0, aP1, bP0, bP1, c00, c01, c10, c11);
  else       gemm_tiles<false>(aP0, aP1, bP0, bP1, c00, c01, c10, c11);

  int col = nBase + l16;
#pragma unroll
  for (int v = 0; v < 8; ++v) {
    int r0 = mBase + v + half * 8;
    ap[(size_t)r0        * BANK_ROWS + col] = c00[v];
    ap[(size_t)(r0 + 16) * BANK_ROWS + col] = c10[v];
    if (fullN) {
      ap[(size_t)r0        * BANK_ROWS + col + 16] = c01[v];
      ap[(size_t)(r0 + 16) * BANK_ROWS + col + 16] = c11[v];
    }
  }
}

// ---------------------------------------------------------------- scatter fixup
// Columns enqueue_idx[n] must reflect bank[idx] := f[n] (last writer wins).
// For the winning n, overwrite all_pairs[:, idx] with f @ f[n].
__global__ __launch_bounds__(256)
void fixup_k(const float* __restrict__ f, const int* __restrict__ enq,
             float* __restrict__ ap) {
  __shared__ float fv[CHANNEL];
  __shared__ int winner;
  int n = blockIdx.x;
  int t = threadIdx.x;
  if (t == 0) {
    int idx = enq[n], w = 1;
    for (int k = n + 1; k < BATCH; ++k)
      if (enq[k] == idx) { w = 0; break; }   // a later write supersedes n
    winner = w;
  }
  __syncthreads();
  if (!winner) return;
  int idx = enq[n];
  fv[t] = f[(size_t)n * CHANNEL + t];
  __syncthreads();
  for (int r = t; r < BATCH; r += 256) {
    const float* fr = f + (size_t)r * CHANNEL;
    float acc = 0.0f;
#pragma unroll 8
    for (int c = 0; c < CHANNEL; ++c) acc += fr[c] * fv[c];
    ap[(size_t)r * BANK_ROWS + idx] = acc;
  }
}

// ---------------------------------------------------------------- radix select
struct SelResult { unsigned thr; int ties_needed; unsigned ties_total; };

// Exact descending top-K threshold over negative columns via 4x 8-bit radix
// rounds on monotonic keys. invert=true selects the K smallest raw values.
__device__ SelResult radix_select_desc(const float* __restrict__ vals,
                                       const float* __restrict__ flagUpd,
                                       int lab, int K, bool invert,
                                       unsigned* hist, volatile int* sh, int tid) {
  unsigned prefixTop = 0, maskTop = 0;
  int rem = K;
  for (int shift = 24; shift >= 0; shift -= 8) {
    for (int i = tid; i < 256; i += 512) hist[i] = 0u;
    __syncthreads();
    for (int m = tid; m < BANK_ROWS; m += 512) {
      if (m % CLASS_NUM == lab) continue;
      if (!(flagUpd[m] > 0.0f)) continue;
      unsigned key = fkey(vals[m]);
      if (invert) key = ~key;
      if ((key & maskTop) != prefixTop) continue;
      atomicAdd(&hist[(key >> shift) & 255u], 1u);
    }
    __syncthreads();
    if (tid == 0) {
      unsigned cum = 0; int b = 255;
      for (; b > 0; --b) {
        if (cum + hist[b] >= (unsigned)rem) break;
        cum += hist[b];
      }
      sh[0] = b; sh[1] = (int)cum; sh[2] = (int)hist[b];
    }
    __syncthreads();
    int b = sh[0];
    rem -= sh[1];
    prefixTop |= ((unsigned)b) << shift;
    maskTop   |= 0xFFu << shift;
    __syncthreads();
  }
  SelResult r;
  r.thr = prefixTop; r.ties_needed = rem; r.ties_total = (unsigned)sh[2];
  return r;
}

// ---------------------------------------------------------------- per-row loss
__global__ __launch_bounds__(512)
void reduce_k(const float* __restrict__ ap, const float* __restrict__ rand_neg,
              const int* __restrict__ label, const float* __restrict__ flagUpd,
              float* __restrict__ out) {
  __shared__ float pos[1024];
  __shared__ unsigned hist[256];
  __shared__ int sh[4];
  __shared__ float red[512];
  __shared__ float shLh, shLr;

  int n   = blockIdx.x;
  int tid = threadIdx.x;
  int lab = label[n];
  const float* row  = ap       + (size_t)n * BANK_ROWS;
  const float* rrow = rand_neg + (size_t)n * BANK_ROWS;

  // ---- gather positives (m % 60 == label), pad to 1024 with +inf ----
  for (int j = tid; j < 1024; j += 512) {
    float v = INFINITY;
    if (j < MEM_SIZE) {
      int m = lab + CLASS_NUM * j;
      if (flagUpd[m] > 0.0f) v = row[m];
    }
    pos[j] = v;
  }
  __syncthreads();
  // bitonic sort ascending, 1024 elems / 512 threads
  for (int k = 2; k <= 1024; k <<= 1) {
    for (int j = k >> 1; j > 0; j >>= 1) {
      int i = ((tid / j) * 2 * j) + (tid % j);
      int p = i + j;
      bool up = ((i & k) == 0);
      float a = pos[i], b = pos[p];
      if (up ? (a > b) : (a < b)) { pos[i] = b; pos[p] = a; }
      __syncthreads();
    }
  }

  // ---- hard negatives: top-512 sims; ties share one value -> exact ----
  SelResult rh = radix_select_desc(row, flagUpd, lab, NEG_NUM, false, hist, sh, tid);
  float part = 0.0f;   // sims in [-1,1] after normalization: exp is safe unshifted
  for (int m = tid; m < BANK_ROWS; m += 512) {
    if (m % CLASS_NUM == lab) continue;
    if (!(flagUpd[m] > 0.0f)) continue;
    if (fkey(row[m]) > rh.thr) part += expf(row[m]);
  }
  red[tid] = part; __syncthreads();
  for (int s = 256; s > 0; s >>= 1) {
    if (tid < s) red[tid] += red[tid + s];
    __syncthreads();
  }
  if (tid == 0) {
    float tot = red[0] + (float)rh.ties_needed * expf(keyToFloat(rh.thr));
    shLh = logf(tot);
  }
  __syncthreads();

  // ---- random negatives: 512 smallest rand values (stable over index) ----
  SelResult rrs = radix_select_desc(rrow, flagUpd, lab, NEG_NUM, true, hist, sh, tid);
  bool allTies = (rrs.ties_total == (unsigned)rrs.ties_needed); // uniform
  part = 0.0f;
  for (int m = tid; m < BANK_ROWS; m += 512) {
    if (m % CLASS_NUM == lab) continue;
    if (!(flagUpd[m] > 0.0f)) continue;
    unsigned key = ~fkey(rrow[m]);
    if (key > rrs.thr || (allTies && key == rrs.thr)) part += expf(row[m]);
  }
  red[tid] = part; __syncthreads();
  for (int s = 256; s > 0; s >>= 1) {
    if (tid < s) red[tid] += red[tid + s];
    __syncthreads();
  }
  if (tid == 0) {
    float tot = red[0];
    if (!allTies) {                      // rare float-duplicate rand: index order
      int need = rrs.ties_needed;
      for (int m = 0; m < BANK_ROWS && need > 0; ++m) {
        if (m % CLASS_NUM == lab) continue;
        if (!(flagUpd[m] > 0.0f)) continue;
        if (~fkey(rrow[m]) == rrs.thr) { tot += expf(row[m]); --need; }
      }
    }
    shLr = logf(tot);
  }
  __syncthreads();

  // ---- per-positive softplus terms, mean-accumulate ----
  float term = 0.0f;
  if (tid < POS_NUM) {
    float p = pos[tid];
    term = log1pf(expf(shLh - p)) + log1pf(expf(shLr - p));
  }
  red[tid] = term; __syncthreads();
  for (int s = 256; s > 0; s >>= 1) {
    if (tid < s) red[tid] += red[tid + s];
    __syncthreads();
  }
  if (tid == 0)
    atomicAdd(out, red[0] * (1.0f / (float)(BATCH * 2 * POS_NUM)));
}

// ---------------------------------------------------------------- launcher
extern "C" void kernel_launch(void* const* d_in, const int* in_sizes, int n_in,
                              void* d_out, int out_size, void* d_ws, size_t ws_size,
                              hipStream_t stream) {
  const float* f_normed  = (const float*)d_in[0];
  const int*   label     = (const int*)  d_in[1];
  const int*   enq       = (const int*)  d_in[2];
  const float* bank      = (const float*)d_in[3];
  const float* bank_flag = (const float*)d_in[4];
  const float* rand_neg  = (const float*)d_in[5];
  float* out = (float*)d_out;

  float* ap      = (float*)d_ws;                       // 512*41040 f32 (~84 MB)
  float* flagUpd = ap + (size_t)BATCH * BANK_ROWS;     // 41040 f32

  zero_out_k<<<1, 1, 0, stream>>>(out);
  flag_init_k<<<(BANK_ROWS + 255) / 256, 256, 0, stream>>>(bank_flag, flagUpd);
  flag_scatter_k<<<1, BATCH, 0, stream>>>(enq, flagUpd);

  int total_waves = MS_TILES * NS_TILES;               // 16 * 1283 = 20528
  gemm_wmma_k<<<total_waves / 8, 256, 0, stream>>>(f_normed, bank, ap);
  fixup_k<<<BATCH, 256, 0, stream>>>(f_normed, enq, ap);
  reduce_k<<<BATCH, 512, 0, stream>>>(ap, rand_neg, label, flagUpd, out);
}